// GatedGCN_73658689126824
// MI455X (gfx1250) — compile-verified
//
#include <hip/hip_runtime.h>
#include <math.h>

// GatedGCN on MI455X (gfx1250, wave32).
// Roofline: edge gather/scatter dominates (~0.5 GB/fwd => ~21us at 23.3 TB/s);
// dense math (~3 GFLOP) rides the f32 WMMA pipe (V_WMMA_F32_16X16X4_F32).
// Scatter uses hardware global_atomic_add_f32 (unsafeAtomicAdd) -- a CAS loop
// here would multiply the dominant traffic term.

typedef __attribute__((ext_vector_type(2))) float v2f;
typedef __attribute__((ext_vector_type(8))) float v8f;

#define H       32
#define NLAYERS 2

// ---------------------------------------------------------------------------
// Y_tile(16x16) += X(16xK, row-major ld=ldx) @ W(rows n=0..15 of length K)^T
// using V_WMMA_F32_16X16X4_F32, K stepped by 4.
// A frag (f32 16x4): lane l holds A[m=l&15][k0+2*(l>>4) + {0,1}]  (one v2f)
// B frag (f32 4x16): lane l holds B[k0+2*(l>>4) + {0,1}][n=l&15] = W[n][k...]
// X/W may live in global or LDS (generic pointers; LDS reads -> ds_load_b64).
// ---------------------------------------------------------------------------
__device__ __forceinline__ v8f wmma_xwt(const float* X, int ldx,
                                        const float* W, int ldw,
                                        int K, int lane, v8f acc) {
    const int m  = lane & 15;          // also n for the B fragment
    const int kh = (lane >> 4) << 1;   // 0 or 2
#pragma unroll
    for (int k0 = 0; k0 < K; k0 += 4) {
        v2f a = *(const v2f*)(X + m * ldx + k0 + kh);
        v2f b = *(const v2f*)(W + m * ldw + k0 + kh);
        acc = __builtin_amdgcn_wmma_f32_16x16x4_f32(
            /*neg_a=*/false, a, /*neg_b=*/false, b,
            /*c_mod=*/(short)0, acc, /*reuse_a=*/false, /*reuse_b=*/false);
    }
    return acc;
}

// C/D fragment layout: vgpr v, lane l -> (m = v + 8*(l>=16), n = l&15)
__device__ __forceinline__ void store_cfrag(float* __restrict__ Y, int ldy,
                                            int lane, v8f c) {
    const int n  = lane & 15;
    const int mh = (lane >> 4) * 8;
#pragma unroll
    for (int v = 0; v < 8; ++v) Y[(mh + v) * ldy + n] = c[v];
}

// ---------------------------------------------------------------------------
// Kernel 1: h = relu(x @ W_first^T + b_first).  One wave per 16-row tile.
// ---------------------------------------------------------------------------
__global__ void first_lin_kernel(const float* __restrict__ x,
                                 const float* __restrict__ Wf,
                                 const float* __restrict__ bf,
                                 float* __restrict__ h, int ntiles) {
    __shared__ float sW[H * H];                      // 4KB, shared by 8 waves
    for (int i = threadIdx.x; i < H * H; i += blockDim.x) sW[i] = Wf[i];
    __syncthreads();

    const int lane = threadIdx.x & 31;
    const int wave = threadIdx.x >> 5;
    const int tile = blockIdx.x * (blockDim.x >> 5) + wave;
    if (tile >= ntiles) return;                      // wave-uniform guard
    const float* X = x + (size_t)tile * 16 * H;
    float* Y       = h + (size_t)tile * 16 * H;
#pragma unroll
    for (int t = 0; t < 2; ++t) {                    // two 16-col output tiles
        v8f acc = {};
        acc = wmma_xwt(X, H, sW + t * 16 * H, H, H, lane, acc);
        const float bias = bf[t * 16 + (lane & 15)];
#pragma unroll
        for (int v = 0; v < 8; ++v) {
            float y = acc[v] + bias;
            acc[v] = y > 0.f ? y : 0.f;              // relu
        }
        store_cfrag(Y + t * 16, H, lane, acc);
    }
}

// ---------------------------------------------------------------------------
// Kernel 2a: zero the aggregation buffer.
// ---------------------------------------------------------------------------
__global__ void zero_kernel(float* __restrict__ p, int n) {
    int i = blockIdx.x * blockDim.x + threadIdx.x;
    if (i < n) p[i] = 0.f;
}

// ---------------------------------------------------------------------------
// Kernel 2b: agg[dst] += edge_weight * h[src].  Lane = feature (coalesced
// 128B row gather); scatter via hardware global_atomic_add_f32.
// ---------------------------------------------------------------------------
__global__ void scatter_kernel(const float* __restrict__ h,
                               const int* __restrict__ src,
                               const int* __restrict__ dst,
                               const float* __restrict__ ew,
                               float* __restrict__ agg, int nedges) {
    long gid = (long)blockIdx.x * blockDim.x + threadIdx.x;
    int e = (int)(gid >> 5);
    int f = (int)(gid & 31);
    if (e >= nedges) return;
    float v = ew[e] * h[(size_t)src[e] * H + f];
    unsafeAtomicAdd(agg + (size_t)dst[e] * H + f, v);
}

// ---------------------------------------------------------------------------
// Kernel 3: fused  out_i = agg@Wnn^T+b ; GRU(out_i, h) ; h = h_new + fuse*h.
// One wave per 16-row tile. All three weight matrices are block-preloaded
// into LDS (8 waves share them -> 8x less global weight traffic); out_i is
// bounced through wave-private LDS so it can be re-read in A-fragment layout
// for the 32->96 GRU GEMMs.
// ---------------------------------------------------------------------------
__global__ void layer_kernel(const float* __restrict__ agg,
                             float* __restrict__ h,
                             const float* __restrict__ Wnn,
                             const float* __restrict__ bnn,
                             const float* __restrict__ Wih,
                             const float* __restrict__ bih,
                             const float* __restrict__ Whh,
                             const float* __restrict__ bhh,
                             const float* __restrict__ fuse_w,
                             int layer, int ntiles) {
    __shared__ float sWnn[H * H];                    //  4KB
    __shared__ float sWih[3 * H * H];                // 12KB
    __shared__ float sWhh[3 * H * H];                // 12KB
    __shared__ float lds_outi[8][16 * H];            // 16KB (2KB per wave)

    // cooperative weight preload (tile-independent; before the guard so all
    // waves reach the barrier)
    for (int i = threadIdx.x; i < H * H; i += blockDim.x)     sWnn[i] = Wnn[i];
    for (int i = threadIdx.x; i < 3 * H * H; i += blockDim.x) sWih[i] = Wih[i];
    for (int i = threadIdx.x; i < 3 * H * H; i += blockDim.x) sWhh[i] = Whh[i];
    __syncthreads();

    const int lane = threadIdx.x & 31;
    const int wave = threadIdx.x >> 5;
    const int tile = blockIdx.x * (blockDim.x >> 5) + wave;
    if (tile >= ntiles) return;                      // wave-uniform guard
    const float* A = agg + (size_t)tile * 16 * H;
    float* Hrow    = h   + (size_t)tile * 16 * H;
    float* S       = lds_outi[wave];
    const float fuse = fuse_w[layer];

    const int n  = lane & 15;
    const int mh = (lane >> 4) * 8;

    // out_i = agg @ Wnn^T + bnn  -> LDS (row-major 16xH)
#pragma unroll
    for (int t = 0; t < 2; ++t) {
        v8f acc = {};
        acc = wmma_xwt(A, H, sWnn + t * 16 * H, H, H, lane, acc);
        const float bias = bnn[t * 16 + n];
#pragma unroll
        for (int v = 0; v < 8; ++v)
            S[(mh + v) * H + t * 16 + n] = acc[v] + bias;
    }
    // wave-private LDS region: DS pipeline is in-order within a wave.

    // gi = out_i @ Wih^T + bih  (6 tiles of 16 cols = 96)
    // gh = h     @ Whh^T + bhh
    v8f gi[6], gh[6];
#pragma unroll
    for (int t = 0; t < 6; ++t) {
        v8f acc = {};
        acc = wmma_xwt(S, H, sWih + t * 16 * H, H, H, lane, acc);
        const float bi = bih[t * 16 + n];
#pragma unroll
        for (int v = 0; v < 8; ++v) acc[v] += bi;
        gi[t] = acc;

        v8f acc2 = {};
        acc2 = wmma_xwt(Hrow, H, sWhh + t * 16 * H, H, H, lane, acc2);
        const float bh = bhh[t * 16 + n];
#pragma unroll
        for (int v = 0; v < 8; ++v) acc2[v] += bh;
        gh[t] = acc2;
    }

    // GRU gates + fuse: tiles {0,1}=r, {2,3}=z, {4,5}=n along the 96 dim.
#pragma unroll
    for (int t = 0; t < 2; ++t) {
#pragma unroll
        for (int v = 0; v < 8; ++v) {
            const int idx = (mh + v) * H + t * 16 + n;
            float xf = Hrow[idx];
            float r  = 1.f / (1.f + __expf(-(gi[t][v]     + gh[t][v])));
            float z  = 1.f / (1.f + __expf(-(gi[2 + t][v] + gh[2 + t][v])));
            float nn = tanhf(gi[4 + t][v] + r * gh[4 + t][v]);
            float hn = (1.f - z) * nn + z * xf;
            Hrow[idx] = hn + fuse * xf;              // in-place: own rows only
        }
    }
}

// ---------------------------------------------------------------------------
// Kernel 4: logits = h @ W_out^T + b_out; 2-class log_softmax. Pure VALU.
// ---------------------------------------------------------------------------
__global__ void head_kernel(const float* __restrict__ h,
                            const float* __restrict__ Wout,
                            const float* __restrict__ bout,
                            float* __restrict__ out, int n) {
    int i = blockIdx.x * blockDim.x + threadIdx.x;
    if (i >= n) return;
    const float* row = h + (size_t)i * H;
    float l0 = bout[0], l1 = bout[1];
#pragma unroll
    for (int k = 0; k < H; ++k) {
        float x = row[k];
        l0 = fmaf(x, Wout[k], l0);
        l1 = fmaf(x, Wout[H + k], l1);
    }
    float m   = fmaxf(l0, l1);
    float lse = m + __logf(__expf(l0 - m) + __expf(l1 - m));
    out[2 * i]     = l0 - lse;
    out[2 * i + 1] = l1 - lse;
}

// ---------------------------------------------------------------------------
extern "C" void kernel_launch(void* const* d_in, const int* in_sizes, int n_in,
                              void* d_out, int out_size, void* d_ws, size_t ws_size,
                              hipStream_t stream) {
    const float* x    = (const float*)d_in[0];
    const int*   ei   = (const int*)  d_in[1];   // [2, E]
    const float* ew   = (const float*)d_in[2];
    const float* Wf   = (const float*)d_in[3];
    const float* bf   = (const float*)d_in[4];
    const float* Wnn  = (const float*)d_in[5];   // [L, H, H]
    const float* bnn  = (const float*)d_in[6];   // [L, H]
    const float* Wih  = (const float*)d_in[7];
    const float* bih  = (const float*)d_in[8];
    const float* Whh  = (const float*)d_in[9];
    const float* bhh  = (const float*)d_in[10];
    const float* fw   = (const float*)d_in[11];
    const float* Wout = (const float*)d_in[12];
    const float* bout = (const float*)d_in[13];

    const int E = in_sizes[2];
    const int N = in_sizes[0] / H;               // N % 16 == 0 (100000)
    const int ntiles = (N + 15) / 16;

    float* h   = (float*)d_ws;                   // [ntiles*16, H]
    float* agg = h + (size_t)ntiles * 16 * H;    // [ntiles*16, H]

    const int wavesPerBlk = 8;                   // 256 threads, wave32
    dim3 blk(256);
    dim3 grid_t((ntiles + wavesPerBlk - 1) / wavesPerBlk);

    first_lin_kernel<<<grid_t, blk, 0, stream>>>(x, Wf, bf, h, ntiles);

    const int* src = ei;
    const int* dst = ei + E;
    const int nAgg = N * H;
    const long scatterThreads = (long)E * 32;
    for (int i = 0; i < NLAYERS; ++i) {
        zero_kernel<<<(nAgg + 255) / 256, 256, 0, stream>>>(agg, nAgg);
        scatter_kernel<<<(unsigned)((scatterThreads + 255) / 256), 256, 0, stream>>>(
            h, src, dst, ew, agg, E);
        layer_kernel<<<grid_t, blk, 0, stream>>>(
            agg, h, Wnn + (size_t)i * H * H, bnn + (size_t)i * H,
            Wih, bih, Whh, bhh, fw, i, ntiles);
    }

    head_kernel<<<(N + 255) / 256, 256, 0, stream>>>(h, Wout, bout,
                                                     (float*)d_out, N);
}